// MeanAggregator_2826088481576
// MI455X (gfx1250) — compile-verified
//
#include <hip/hip_runtime.h>

// MeanAggregator for MI455X (gfx1250):
//   out[b, :] = (1/32) * sum_k embed[neigh_ids[b,k], :]
// Memory-bound random row gather (51.2 MB table -> L2-resident on 192MB L2).
// Strategy: Tensor Data Mover gather-mode DMA (8 rows / instruction) into LDS,
// TENSORcnt-pipelined drain, f32 accumulate with ds_load_b128, b128 store.

typedef float        v4f __attribute__((ext_vector_type(4)));
typedef unsigned int v4u __attribute__((ext_vector_type(4)));
typedef int          v8i __attribute__((ext_vector_type(8)));
typedef int          v4i __attribute__((ext_vector_type(4)));

#define NUM_V 100000
#define DIM   128
#define NB    8192
#define KNE   32
#define WAVES_PER_BLOCK 4   // 128 threads; 64 KB LDS / block

__global__ __launch_bounds__(WAVES_PER_BLOCK * 32, 1)
void MeanAggregator_tdm_kernel(const int* __restrict__ neigh_ids,
                               const float* __restrict__ embed,
                               float* __restrict__ out)
{
    // Per-wave staging buffer: 32 rows x 128 f32 = 16 KB.
    __shared__ float lds[WAVES_PER_BLOCK * KNE * DIM];   // 64 KB static LDS

    const int lane = threadIdx.x & 31;
    const int wave = threadIdx.x >> 5;
    const int row  = blockIdx.x * WAVES_PER_BLOCK + wave; // one output row / wave
    if (row >= NB) return;                                // wave-uniform guard

    // Wave-uniform LDS byte offset of this wave's staging buffer.
    // Generic-pointer low 32 bits == workgroup-relative LDS byte address.
    unsigned ldsBase = (unsigned)(unsigned long long)(&lds[wave * KNE * DIM]);
    ldsBase = (unsigned)__builtin_amdgcn_readfirstlane((int)ldsBase);

    // One coalesced b32 load: lane L holds neighbor id L for this row.
    const int myId = neigh_ids[row * KNE + lane];

    // ---- Tensor DMA descriptor (gather mode, 32-bit indices, 8 rows/op) ----
    const unsigned long long gbase = (unsigned long long)embed;

    v4u g0;
    g0.x = 0xC0000001u;                               // count=1 | idx32(b30) | gather(b31)
    g0.y = 0;                                         // lds_addr (set per batch below)
    g0.z = (unsigned)(gbase & 0xFFFFFFFFull);         // global_addr[31:0]
    g0.w = (unsigned)((gbase >> 32) & 0x01FFFFFFull)  // global_addr[56:32]
         | 0x80000000u;                               // type = 2 ("image")

    v8i g1;
    g1[0] = (int)(2u << 16);                              // data_size = 4 bytes
    g1[1] = (int)(((unsigned)DIM   & 0xFFFFu) << 16);     // tensor_dim0[15:0] = 128
    g1[2] = (int)(((unsigned)NUM_V & 0xFFFFu) << 16);     // dim0[31:16]=0 | tensor_dim1 lo
    g1[3] = (int)(((unsigned)DIM << 16)                   // tile_dim0 = 128
                | ((unsigned)NUM_V >> 16));               // tensor_dim1 hi
    g1[4] = 8;                                            // tile_dim1 = 8 valid indices
    g1[5] = DIM;                                          // tensor_dim0_stride[31:0] = 128
    g1[6] = 0;                                            // stride hi | dim1_stride (ignored)
    g1[7] = 0;

    // Issue 4 async gathers: batch g pulls neighbor rows [8g, 8g+8) into
    // lds[ldsBase + g*4096 ...]. EXEC-independent, tracked by TENSORcnt.
    #pragma unroll
    for (int g = 0; g < 4; ++g) {
        g0.y = ldsBase + (unsigned)g * (8u * DIM * 4u);
        v4i g2, g3;
        #pragma unroll
        for (int j = 0; j < 4; ++j) {
            g2[j] = __builtin_amdgcn_readlane(myId, g * 8 + j);
            g3[j] = __builtin_amdgcn_readlane(myId, g * 8 + 4 + j);
        }
        // 6-arg toolchain form: trailing int32x8 is uncharacterized; mirror the
        // gather indices (group2||group3) into it so either operand layout sees
        // the same 8 row ids. cpol = 0.
        v8i gx;
        #pragma unroll
        for (int j = 0; j < 4; ++j) { gx[j] = g2[j]; gx[j + 4] = g3[j]; }
        __builtin_amdgcn_tensor_load_to_lds(g0, g1, g2, g3, gx, 0);
    }

    // ---- Pipelined drain: wait tensorcnt<=3,2,1,0 and accumulate ----
    const float* myLds = &lds[wave * KNE * DIM];
    v4f acc = (v4f)0.0f;

#define DRAIN_BATCH(G, CNT)                                                    \
    do {                                                                       \
        __builtin_amdgcn_s_wait_tensorcnt(CNT);                                \
        asm volatile("" ::: "memory"); /* keep ds reads below the wait */      \
        _Pragma("unroll")                                                      \
        for (int k = 0; k < 8; ++k) {                                          \
            /* ds_load_b128, conflict-free: byte addr = row*512 + lane*16 */   \
            const v4f v = *(const v4f*)(myLds + ((G) * 8 + k) * DIM + lane * 4); \
            acc += v;                                                          \
        }                                                                      \
    } while (0)

    DRAIN_BATCH(0, 3);
    DRAIN_BATCH(1, 2);
    DRAIN_BATCH(2, 1);
    DRAIN_BATCH(3, 0);
#undef DRAIN_BATCH

    acc *= (1.0f / (float)KNE);
    *(v4f*)(out + (unsigned long long)row * DIM + lane * 4) = acc;  // b128 store
}

extern "C" void kernel_launch(void* const* d_in, const int* in_sizes, int n_in,
                              void* d_out, int out_size, void* d_ws, size_t ws_size,
                              hipStream_t stream) {
    const int*   neigh = (const int*)d_in[0];     // [B, K] int32 ids
    const float* embed = (const float*)d_in[1];   // [V, D] f32 table
    float*       outp  = (float*)d_out;           // [B, D] f32
    (void)in_sizes; (void)n_in; (void)out_size; (void)d_ws; (void)ws_size;

    dim3 grid(NB / WAVES_PER_BLOCK);   // 2048 blocks
    dim3 block(WAVES_PER_BLOCK * 32);  // 128 threads = 4 waves
    hipLaunchKernelGGL(MeanAggregator_tdm_kernel, grid, block, 0, stream,
                       neigh, embed, outp);
}